// TorchDCTExtractor_76192719831840
// MI455X (gfx1250) — compile-verified
//
#include <hip/hip_runtime.h>
#include <hip/hip_bf16.h>
#include <math.h>

// ---------------------------------------------------------------------------
// MI455X (gfx1250) single-pass DCT band-feature extractor.
// Memory-bound: 402MB input @ 23.3TB/s => ~17us floor. One fused pass:
//   RGB -> lum (track global max) -> 8x8 DCT via V_WMMA_F32_16X16X4_F32
//   (4 blocks per 16x16 tile using block-diagonal DCT operand)
//   -> per-r |coeff| running sums in registers (band grouping is
//      loop-invariant, applied once per wave) -> tiny finalize kernel.
// ---------------------------------------------------------------------------

typedef __attribute__((ext_vector_type(2))) float v2f;
typedef __attribute__((ext_vector_type(4))) float v4f;
typedef __attribute__((ext_vector_type(8))) float v8f;

#define IMG_H 1024
#define IMG_W 1024
#define N_IMG 32
#define N_BAND 8
#define NH (IMG_H / 8)           // 128 blocks per column
#define NW (IMG_W / 8)           // 128 blocks per row
#define TILES_PER_IMG ((NH / 2) * (NW / 2))   // 4096 16x16 tiles
#define WGS_PER_IMG 32
#define TILES_PER_WG (TILES_PER_IMG / WGS_PER_IMG)   // 128
#define WAVES_PER_WG 8
#define TILES_PER_WAVE (TILES_PER_WG / WAVES_PER_WG) // 16
#define LDS_STRIDE 20            // dwords; 20*8 mod 64 == 32 -> lane-half banks disjoint
#define STAT_N ((float)(NH * NW * 8))  // 131072 samples per (image, band)

// Build inverse-zigzag band table: band_of[row*8+col] = zigzag_position >> 3
__global__ void build_tables_kernel(const int* __restrict__ zz, int* __restrict__ band_of) {
    int p = threadIdx.x;
    if (p < 64) band_of[zz[p]] = p >> 3;
}

__global__ __launch_bounds__(256)
void dct_stats_kernel(const float* __restrict__ img,
                      const float* __restrict__ dmat,      // 8x8 DCT matrix
                      const int*   __restrict__ band_of_g, // 64-entry band table
                      float* __restrict__ sums,            // [32][8]
                      float* __restrict__ sqs,             // [32][8]
                      unsigned* __restrict__ maxbits)      // global max(lum) as uint bits
{
    __shared__ float tile[WAVES_PER_WG][16 * LDS_STRIDE];
    __shared__ float Dlds[64];
    __shared__ int   bandLds[64];
    __shared__ float wgSum[N_BAND], wgSq[N_BAND];
    __shared__ unsigned wgMax;

    const int tid = threadIdx.x;
    if (tid < 64) { Dlds[tid] = dmat[tid]; bandLds[tid] = band_of_g[tid]; }
    if (tid < N_BAND) { wgSum[tid] = 0.f; wgSq[tid] = 0.f; }
    if (tid == 16) wgMax = 0u;
    __syncthreads();

    const int lane    = tid & 31;
    const int wave    = tid >> 5;
    const int image   = blockIdx.x >> 5;    // 32 WGs per image
    const int wgInImg = blockIdx.x & 31;

    const int  m  = lane & 15;
    const bool hi = lane >= 16;

    // Constant WMMA operand: Dbd = diag(D, D), 16x16, sliced into 4 K-chunks of 4.
    // A layout (16x4 f32): lanes0-15 hold (M=lane, K=4kk,4kk+1); lanes16-31 (M=lane-16, K=4kk+2,4kk+3).
    // Under the mirrored B layout, the same registers encode diag(D^T,D^T) as a B operand.
    v2f dAB[4];
#pragma unroll
    for (int kk = 0; kk < 4; ++kk) {
        int k0 = 4 * kk + (hi ? 2 : 0);      // k0 even; k0 and k0+1 share a quadrant
        bool mh = m >= 8, kh = k0 >= 8;
        if (mh == kh) {
            int mm = m & 7, kq = k0 & 7;
            dAB[kk][0] = Dlds[mm * 8 + kq];
            dAB[kk][1] = Dlds[mm * 8 + kq + 1];
        } else {
            dAB[kk][0] = 0.f;
            dAB[kk][1] = 0.f;
        }
    }

    // Loop-invariant per-lane band ids: C/D layout VGPR r holds element
    // (M = r (+8), N = lane&15) -> intra-block coords (i = r, j = lane&7).
    int bandv[8];
#pragma unroll
    for (int r = 0; r < 8; ++r) bandv[r] = bandLds[r * 8 + (lane & 7)];

    // Per-r running statistics (band grouping deferred to the epilogue).
    float sR[8], qR[8];
#pragma unroll
    for (int r = 0; r < 8; ++r) { sR[r] = 0.f; qR[r] = 0.f; }
    float maxv = 0.f;

    float* my = tile[wave];
    const float* base = img + (size_t)image * 3 * IMG_H * IMG_W;

#pragma unroll 1
    for (int it = 0; it < TILES_PER_WAVE; ++it) {
        const int t    = wgInImg * TILES_PER_WG + wave * TILES_PER_WAVE + it;
        const int ty   = t >> 6, tx = t & 63;
        const int row0 = ty * 16, col0 = tx * 16;

        // ---- 16x16 luminance tile -> LDS (coalesced b128 loads, 4 lanes/row) ----
#pragma unroll
        for (int half = 0; half < 2; ++half) {
            int row = (lane >> 2) + half * 8;
            int col = (lane & 3) * 4;
            size_t off = (size_t)(row0 + row) * IMG_W + (col0 + col);
            v4f r4 = *(const v4f*)(base + off);
            v4f g4 = *(const v4f*)(base + (size_t)IMG_H * IMG_W + off);
            v4f b4 = *(const v4f*)(base + (size_t)2 * IMG_H * IMG_W + off);
            if (it + 1 < TILES_PER_WAVE) {   // prefetch next tile (stays in-image)
                __builtin_prefetch(base + off + 16, 0, 1);
                __builtin_prefetch(base + (size_t)IMG_H * IMG_W + off + 16, 0, 1);
                __builtin_prefetch(base + (size_t)2 * IMG_H * IMG_W + off + 16, 0, 1);
            }
            v4f l4 = 0.299f * r4 + 0.587f * g4 + 0.114f * b4;
            maxv = fmaxf(maxv, fmaxf(fmaxf(l4[0], l4[1]), fmaxf(l4[2], l4[3])));
            *(v4f*)(my + row * LDS_STRIDE + col) = l4;
        }
        asm volatile("s_wait_dscnt 0" ::: "memory");

        // ---- stage 1: T = diag(D,D) @ X  (4 chained f32 WMMAs, K=16) ----
        v8f c = {};
#pragma unroll
        for (int kk = 0; kk < 4; ++kk) {
            int k0 = 4 * kk + (hi ? 2 : 0);
            v2f bx;
            bx[0] = my[k0 * LDS_STRIDE + m];          // B[k0][n],   n = lane&15
            bx[1] = my[(k0 + 1) * LDS_STRIDE + m];    // B[k0+1][n]
            c = __builtin_amdgcn_wmma_f32_16x16x4_f32(false, dAB[kk], false, bx,
                                                      (short)0, c, false, false);
        }

        // ---- C/D layout -> A layout via LDS bounce (reuse tile buffer) ----
#pragma unroll
        for (int r = 0; r < 8; ++r) {
            int M = r + (hi ? 8 : 0);
            my[M * LDS_STRIDE + m] = c[r];
        }
        asm volatile("s_wait_dscnt 0" ::: "memory");

        v2f aT[4];
#pragma unroll
        for (int kk = 0; kk < 4; ++kk) {
            int k0 = 4 * kk + (hi ? 2 : 0);
            aT[kk][0] = my[m * LDS_STRIDE + k0];
            aT[kk][1] = my[m * LDS_STRIDE + k0 + 1];
        }

        // ---- stage 2: Y = T @ diag(D^T,D^T) ----
        v8f y = {};
#pragma unroll
        for (int kk = 0; kk < 4; ++kk)
            y = __builtin_amdgcn_wmma_f32_16x16x4_f32(false, aT[kk], false, dAB[kk],
                                                      (short)0, y, false, false);

        // ---- cheap per-r accumulation (abs folds into src modifiers) ----
#pragma unroll
        for (int r = 0; r < 8; ++r) {
            float av = fabsf(y[r]);
            sR[r] += av;
            qR[r] = fmaf(av, av, qR[r]);
        }
    }

    // ---- once per wave: regroup per-r sums into bands ----
    float accS[N_BAND], accQ[N_BAND];
#pragma unroll
    for (int b = 0; b < N_BAND; ++b) {
        float s = 0.f, q = 0.f;
#pragma unroll
        for (int r = 0; r < 8; ++r)
            if (bandv[r] == b) { s += sR[r]; q += qR[r]; }
        accS[b] = s; accQ[b] = q;
    }

    // ---- wave32 shfl-xor reduction, then LDS -> global atomics ----
#pragma unroll
    for (int off = 16; off >= 1; off >>= 1) {
#pragma unroll
        for (int b = 0; b < N_BAND; ++b) {
            accS[b] += __shfl_xor(accS[b], off, 32);
            accQ[b] += __shfl_xor(accQ[b], off, 32);
        }
        maxv = fmaxf(maxv, __shfl_xor(maxv, off, 32));
    }
    if (lane == 0) {
#pragma unroll
        for (int b = 0; b < N_BAND; ++b) {
            atomicAdd(&wgSum[b], accS[b]);
            atomicAdd(&wgSq[b], accQ[b]);
        }
        atomicMax(&wgMax, __float_as_uint(maxv));   // lum >= 0: bits order-preserving
    }
    __syncthreads();
    if (tid < 8)        atomicAdd(&sums[image * 8 + tid], wgSum[tid]);
    else if (tid < 16)  atomicAdd(&sqs[image * 8 + (tid - 8)], wgSq[tid - 8]);
    else if (tid == 16) atomicMax(maxbits, wgMax);
}

__global__ __launch_bounds__(512)
void finalize_kernel(const float* __restrict__ sums, const float* __restrict__ sqs,
                     const unsigned* __restrict__ maxbits,
                     const float* __restrict__ projw,   // [512][16]
                     const float* __restrict__ projb,   // [512]
                     float* __restrict__ out)           // [32][512]
{
    __shared__ float raw[16];
    const int b = blockIdx.x;
    const int f = threadIdx.x;
    if (f < 16) {
        // scale is homogeneous through |.|, mean, std -> apply at the end
        float scale = (__uint_as_float(*maxbits) <= 1.0f) ? 255.f : 1.f;
        int k = f & 7;
        float s = sums[b * 8 + k];
        if (f < 8) {
            raw[f] = (s / STAT_N) * scale;
        } else {
            float q   = sqs[b * 8 + k];
            float var = (q - s * s / STAT_N) / (STAT_N - 1.f);   // ddof=1
            raw[f] = sqrtf(fmaxf(var, 0.f)) * scale;
        }
    }
    __syncthreads();
    float acc = projb[f];
#pragma unroll
    for (int k = 0; k < 16; ++k) acc += raw[k] * projw[f * 16 + k];
    out[b * 512 + f] = acc;
}

extern "C" void kernel_launch(void* const* d_in, const int* in_sizes, int n_in,
                              void* d_out, int out_size, void* d_ws, size_t ws_size,
                              hipStream_t stream) {
    (void)in_sizes; (void)n_in; (void)out_size; (void)ws_size;
    const float* images = (const float*)d_in[0];
    const float* dmat   = (const float*)d_in[1];
    const int*   zz     = (const int*)d_in[2];
    const float* projw  = (const float*)d_in[3];
    const float* projb  = (const float*)d_in[4];
    float* out = (float*)d_out;

    // workspace layout: sums[32][8] | sqs[32][8] | maxbits | band_of[64]
    float*    sums    = (float*)d_ws;
    float*    sqs     = sums + N_IMG * N_BAND;
    unsigned* maxbits = (unsigned*)(sqs + N_IMG * N_BAND);
    int*      band_of = (int*)(maxbits + 1);

    // accumulators must be zero each call (harness does not re-poison)
    hipMemsetAsync(d_ws, 0, (size_t)(2 * N_IMG * N_BAND) * sizeof(float) + sizeof(unsigned),
                   stream);
    build_tables_kernel<<<1, 64, 0, stream>>>(zz, band_of);
    dct_stats_kernel<<<N_IMG * WGS_PER_IMG, 256, 0, stream>>>(images, dmat, band_of,
                                                              sums, sqs, maxbits);
    finalize_kernel<<<N_IMG, 512, 0, stream>>>(sums, sqs, maxbits, projw, projb, out);
}